// EdgestoInter_27504970564309
// MI455X (gfx1250) — compile-verified
//
#include <hip/hip_runtime.h>

// Problem constants (from reference)
#define MM 257
#define NN 257
#define Kn ((MM - 1) * (NN - 1))            // 65536
#define En (MM * (NN - 1) + NN * (MM - 1))  // 131584
#define VBASE (MM * (NN - 1))               // 65792
#define TK 16                               // faces (k) per block

typedef __attribute__((ext_vector_type(2))) float v2f;
typedef __attribute__((ext_vector_type(8))) float v8f;

__global__ __launch_bounds__(512) void edges2faces_wmma(
    const float* __restrict__ x,     // (32,16,E)
    const float* __restrict__ w,     // (16,16,4K)
    const float* __restrict__ bias,  // (16)
    float* __restrict__ out)         // (32,16,K)
{
  // LDS staging; identical [16][32][20] shapes so lane-half base select is one cndmask
  __shared__ float xA[16][32][20];        // t=0 range: e in [k0, k0+16)
  __shared__ float xB[16][32][20];        // t=1 range: e in [k0+256, k0+272)
  __shared__ float xC[16][32][20];        // t=2,3 range: e in [v0, v0+17)
  __shared__ float wb[TK][16][16][4];     // [jl][ic][o][t] -> bank-perfect B reads
  __shared__ float ob[512][17];           // [(b*16+o)][jl] output staging

  const int tid  = threadIdx.x;
  const int blk  = blockIdx.x;
  const int irow = blk >> 4;              // k row (i)
  const int j0   = (blk & 15) * TK;       // k col base (j)
  const int k0   = irow * 256 + j0;
  const long eH0 = (long)k0;              // horizontal edges, row i
  const long eH1 = (long)k0 + 256;        // horizontal edges, row i+1 (L2 reuse!)
  const long eV  = (long)VBASE + (long)irow * 257 + j0; // vertical edges

  // ---- Phase 1: cooperative, coalesced global -> LDS ----
  // xA/xB: 512 pairs (q=b*16+ic) x 16 floats, via float4 (16B-aligned both sides)
  #pragma unroll
  for (int it = 0; it < 4; ++it) {
    int idx4 = it * 512 + tid;            // 0..2047
    int q  = idx4 >> 2;                   // q = b*16 + ic
    int c4 = (idx4 & 3) << 2;
    int b = q >> 4, g = q & 15;
    float4 va = *(const float4*)(x + (long)q * En + eH0 + c4);
    float4 vb = *(const float4*)(x + (long)q * En + eH1 + c4);
    *(float4*)&xA[g][b][c4] = va;
    *(float4*)&xB[g][b][c4] = vb;
  }
  // xC: 512 pairs x 17 floats (v-base odd-strided -> scalar loads)
  #pragma unroll
  for (int it = 0; it < 17; ++it) {
    int idx = it * 512 + tid;             // 0..8703
    int q   = idx / 17;
    int col = idx - q * 17;
    float v = x[(long)q * En + eV + col];
    xC[q & 15][q >> 4][col] = v;
  }
  // Weights: 256 runs (r=o*16+ic) x 64 contiguous floats; read-once -> non-temporal
  const long wk0 = (long)4 * k0;
  #pragma unroll
  for (int it = 0; it < 32; ++it) {
    int idx = it * 512 + tid;             // 0..16383
    int r = idx >> 6;                     // r = o*16 + ic
    int u = idx & 63;                     // u = 4*jl + t
    int o = r >> 4, ic = r & 15;
    int jl = u >> 2, t = u & 3;
    float v = __builtin_nontemporal_load(w + (long)r * (4 * Kn) + wk0 + u);
    wb[jl][ic][o][t] = v;
  }
  __syncthreads();

  // ---- Phase 2: per-wave WMMA, one k per wave ----
  const int wave = tid >> 5;              // jl = wave id (k = k0 + wave)
  const int lane = tid & 31;
  const int m    = lane & 15;             // A: batch row in half; B/D: o = N
  const int s    = lane >> 4;             // lane-half select

  // Hoisted lane-half base selects (one cndmask each); after this every LDS
  // access in the loop is base + compile-time immediate offset.
  const float* aptr0 = (s ? &xC[0][0][0] : &xA[0][0][0]) + wave     + m * 20;
  const float* aptr1 = (s ? &xC[0][0][1] : &xB[0][0][0]) + wave     + m * 20;
  const float* wptr  = &wb[wave][0][m][2 * s];

  float bv = bias[m];                     // D column N = o = m
  v8f acc0 = {bv, bv, bv, bv, bv, bv, bv, bv};  // batches 0..15
  v8f acc1 = acc0;                              // batches 16..31

  #pragma unroll
  for (int g = 0; g < 16; ++g) {          // g = ic chunk, contraction K=4 each
    v2f bf = *(const v2f*)(wptr + g * 64);        // 8B-aligned, bank-perfect
    v2f a0; a0.x = aptr0[g * 640];       a0.y = aptr1[g * 640];        // b = m
    v2f a1; a1.x = aptr0[g * 640 + 320]; a1.y = aptr1[g * 640 + 320]; // b = 16+m
    acc0 = __builtin_amdgcn_wmma_f32_16x16x4_f32(false, a0, false, bf,
                                                 (short)0, acc0, false, false);
    acc1 = __builtin_amdgcn_wmma_f32_16x16x4_f32(false, a1, false, bf,
                                                 (short)0, acc1, false, false);
  }

  // Stage D to LDS: VGPR r -> M = r + 8*s; N = m (all immediate offsets)
  float* obp = &ob[8 * s * 16 + m][wave];
  #pragma unroll
  for (int r = 0; r < 8; ++r) {
    obp[r * 272]        = acc0[r];        // (r*16)*17 words
    obp[r * 272 + 4352] = acc1[r];        // +256*17 words (b += 16)
  }
  __syncthreads();

  // ---- Phase 3: coalesced non-temporal writeout, 64B runs along k ----
  {
    int pr0 = wave * 2 + s;               // pair = b*16 + o
    const float* obr = &ob[pr0][m];
    float* optr = out + (long)pr0 * Kn + k0 + m;
    #pragma unroll
    for (int it = 0; it < 16; ++it) {
      __builtin_nontemporal_store(obr[it * 544], optr); // +32*17 words in LDS
      optr += 32L * Kn;
    }
  }
}

extern "C" void kernel_launch(void* const* d_in, const int* in_sizes, int n_in,
                              void* d_out, int out_size, void* d_ws, size_t ws_size,
                              hipStream_t stream) {
  (void)in_sizes; (void)n_in; (void)out_size; (void)d_ws; (void)ws_size;
  const float* x    = (const float*)d_in[0];
  const float* w    = (const float*)d_in[1];
  const float* bias = (const float*)d_in[2];
  float* out        = (float*)d_out;
  dim3 grid(Kn / TK);   // 4096 blocks, 16 consecutive k each (row-aligned)
  dim3 block(512);      // 16 waves (wave32), one k per wave
  hipLaunchKernelGGL(edges2faces_wmma, grid, block, 0, stream, x, w, bias, out);
}